// myBlur_57440892617279
// MI455X (gfx1250) — compile-verified
//
#include <hip/hip_runtime.h>

// -------- CDNA5 WMMA types (wave32, 16x16 tile striped across the wave) -----
typedef __attribute__((ext_vector_type(2))) float v2f;  // fp32 A/B frag: 16x4 / 4x16
typedef __attribute__((ext_vector_type(8))) float v8f;  // fp32 C/D frag: 16x16

#define KSIZE    121
#define RPAD     60          // (K-1)/2
#define IMW      512
#define IMH      512
#define NPLANE   24          // 8 batch * 3 channels (depthwise => independent planes)
#define KCHUNKS  34          // ceil((121 + 15)/4): banded-Toeplitz contraction length
#define WPAD_LEN 160         // 16 leading zeros + 121 weights + trailing zeros
#define TILES    4           // 16x16 tiles per wave (share weight fragments)

// ---------------------------------------------------------------------------
// Normalized 1D Gaussian from sigma (s = sigma*8+16, var = s^2). Product of
// two normalized 1D kernels == reference's normalized 2D kernel.
// ---------------------------------------------------------------------------
__global__ void gauss_weights_kernel(const float* __restrict__ sigma,
                                     float* __restrict__ d_w) {
    __shared__ float e[KSIZE];
    __shared__ float rsum;
    const int i = threadIdx.x;
    const float s   = sigma[0] * 8.0f + 16.0f;
    const float var = s * s;
    if (i < KSIZE) {
        const float c = (float)i - (float)RPAD;
        e[i] = expf(-(c * c) / (2.0f * var));
    }
    __syncthreads();
    if (i == 0) {
        float acc = 0.0f;
        for (int t = 0; t < KSIZE; ++t) acc += e[t];
        rsum = 1.0f / acc;
    }
    __syncthreads();
    if (i < KSIZE) d_w[i] = e[i] * rsum;
}

__device__ __forceinline__ void load_wpad(const float* __restrict__ wg,
                                          float* __restrict__ wpad) {
    const int lane = threadIdx.x;
    for (int idx = lane; idx < WPAD_LEN; idx += 32) {
        const int t = idx - 16;
        wpad[idx] = (t >= 0 && t < KSIZE) ? wg[t] : 0.0f;
    }
    __syncthreads();
}

__device__ __forceinline__ v8f wmma_f32(v2f a, v2f b, v8f c) {
    return __builtin_amdgcn_wmma_f32_16x16x4_f32(
        /*neg_a=*/false, a, /*neg_b=*/false, b,
        /*c_mod=*/(short)0, c, /*reuse_a=*/false, /*reuse_b=*/false);
}

// ---------------------------------------------------------------------------
// Horizontal pass: 4 tiles stacked in y share the weight (B) fragment.
//   D[y][n] = sum_k A[y][k]*B[k][n],  A[y][k] = x[y0+y][x0+k-60] (clamped),
//   B[k][n] = wpad[16+k-n].
// CLAMP=false: window fully interior -> no clamp math, b64 paired A loads.
// ---------------------------------------------------------------------------
template <bool CLAMP>
__device__ __forceinline__ void hblur_body(const float* __restrict__ xp,
                                           float* __restrict__ op,
                                           const float* __restrict__ wpad,
                                           int x0, int y0) {
    const int lane = threadIdx.x;
    const int half = lane >> 4;   // 0 | 1
    const int lm   = lane & 15;   // M for A, N for B/D

    v8f acc[TILES] = {};
    for (int c = 0; c < KCHUNKS; ++c) {
        const int k0 = 4 * c + 2 * half;
        v2f b;                               // shared weight fragment
        b.x = wpad[16 + k0 - lm];
        b.y = wpad[17 + k0 - lm];
        if (CLAMP) {
            const int xa0 = min(max(x0 + k0 - RPAD, 0), IMW - 1);
            const int xa1 = min(max(x0 + k0 - RPAD + 1, 0), IMW - 1);
#pragma unroll
            for (int t = 0; t < TILES; ++t) {
                const int row = y0 + 16 * t + lm;
                v2f a;
                a.x = xp[row * IMW + xa0];
                a.y = xp[row * IMW + xa1];
                acc[t] = wmma_f32(a, b, acc[t]);
            }
        } else {
            const int xa = x0 + k0 - RPAD;   // in-range; even -> 8B aligned
#pragma unroll
            for (int t = 0; t < TILES; ++t) {
                const int row = y0 + 16 * t + lm;
                const v2f a = *(const v2f*)(xp + row * IMW + xa);
                acc[t] = wmma_f32(a, b, acc[t]);
            }
        }
    }
#pragma unroll
    for (int t = 0; t < TILES; ++t)
#pragma unroll
        for (int v = 0; v < 8; ++v)
            op[(y0 + 16 * t + v + 8 * half) * IMW + (x0 + lm)] = acc[t][v];
}

__global__ __launch_bounds__(32) void hblur_kernel(const float* __restrict__ x,
                                                   const float* __restrict__ wg,
                                                   float* __restrict__ out) {
    __shared__ float wpad[WPAD_LEN];
    load_wpad(wg, wpad);
    const int x0    = blockIdx.x * 16;
    const int y0    = blockIdx.y * (16 * TILES);
    const float* xp = x   + (size_t)blockIdx.z * IMW * IMH;
    float*       op = out + (size_t)blockIdx.z * IMW * IMH;
    // window [x0-60, x0+75] interior iff blockIdx.x in [4,27] (uniform branch)
    if (blockIdx.x >= 4 && blockIdx.x <= 27)
        hblur_body<false>(xp, op, wpad, x0, y0);
    else
        hblur_body<true>(xp, op, wpad, x0, y0);
}

// ---------------------------------------------------------------------------
// Vertical pass: 4 tiles side-by-side in x share the weight (A) fragment.
//   D[m][n] = sum_k A[m][k]*B[k][n],  A[m][k] = wpad[16+k-m],
//   B[k][n] = mid[y0+k-60][x0+n] (clamped).
// ---------------------------------------------------------------------------
template <bool CLAMP>
__device__ __forceinline__ void vblur_body(const float* __restrict__ hp,
                                           float* __restrict__ op,
                                           const float* __restrict__ wpad,
                                           int x0, int y0) {
    const int lane = threadIdx.x;
    const int half = lane >> 4;
    const int lm   = lane & 15;

    v8f acc[TILES] = {};
    for (int c = 0; c < KCHUNKS; ++c) {
        const int k0 = 4 * c + 2 * half;
        v2f a;                               // shared weight fragment
        a.x = wpad[16 + k0 - lm];
        a.y = wpad[17 + k0 - lm];
        int yb0 = y0 + k0 - RPAD;
        int yb1 = yb0 + 1;
        if (CLAMP) {
            yb0 = min(max(yb0, 0), IMH - 1);
            yb1 = min(max(yb1, 0), IMH - 1);
        }
#pragma unroll
        for (int t = 0; t < TILES; ++t) {
            const int col = x0 + 16 * t + lm;
            v2f b;
            b.x = hp[yb0 * IMW + col];
            b.y = hp[yb1 * IMW + col];
            acc[t] = wmma_f32(a, b, acc[t]);
        }
    }
#pragma unroll
    for (int t = 0; t < TILES; ++t)
#pragma unroll
        for (int v = 0; v < 8; ++v)
            op[(y0 + v + 8 * half) * IMW + (x0 + 16 * t + lm)] = acc[t][v];
}

__global__ __launch_bounds__(32) void vblur_kernel(const float* __restrict__ mid,
                                                   const float* __restrict__ wg,
                                                   float* __restrict__ out) {
    __shared__ float wpad[WPAD_LEN];
    load_wpad(wg, wpad);
    const int x0    = blockIdx.x * (16 * TILES);
    const int y0    = blockIdx.y * 16;
    const float* hp = mid + (size_t)blockIdx.z * IMW * IMH;
    float*       op = out + (size_t)blockIdx.z * IMW * IMH;
    // rows [y0-60, y0+75] interior iff blockIdx.y in [4,27] (uniform branch)
    if (blockIdx.y >= 4 && blockIdx.y <= 27)
        vblur_body<false>(hp, op, wpad, x0, y0);
    else
        vblur_body<true>(hp, op, wpad, x0, y0);
}

// ---------------------------------------------------------------------------
// d_in[0] = x (8*3*512*512 fp32), d_in[1] = sigma (1 fp32)
// d_out   = blurred output (fp32, same shape)
// d_ws    = [0..255]: 1D weights; [256..]: horizontal-pass intermediate
//           (~25.2 MB total; intermediate stays resident in the 192 MB L2)
// ---------------------------------------------------------------------------
extern "C" void kernel_launch(void* const* d_in, const int* in_sizes, int n_in,
                              void* d_out, int out_size, void* d_ws, size_t ws_size,
                              hipStream_t stream) {
    (void)in_sizes; (void)n_in; (void)out_size; (void)ws_size;
    const float* x     = (const float*)d_in[0];
    const float* sigma = (const float*)d_in[1];
    float* out  = (float*)d_out;
    float* wbuf = (float*)d_ws;
    float* hmid = (float*)d_ws + 256;

    hipLaunchKernelGGL(gauss_weights_kernel, dim3(1), dim3(128), 0, stream,
                       sigma, wbuf);

    hipLaunchKernelGGL(hblur_kernel, dim3(IMW / 16, IMH / (16 * TILES), NPLANE),
                       dim3(32), 0, stream, x, wbuf, hmid);
    hipLaunchKernelGGL(vblur_kernel, dim3(IMW / (16 * TILES), IMH / 16, NPLANE),
                       dim3(32), 0, stream, hmid, wbuf, out);
}